// PackedSirenExperts_75256416960833
// MI455X (gfx1250) — compile-verified
//
#include <hip/hip_runtime.h>
#include <math.h>

// Problem constants (from reference)
#define B_TOTAL   8192
#define M_EXPERTS 64
#define IN_DIM    3
#define HID       256
#define OUT_DIM   3
#define W0S       30.0f

// Tiling
#define ROWS   32     // batch rows per workgroup (16 per wave)
#define BLOCK  64     // 2 waves (wave32)
#define KBLK   8      // K-panel of W^T staged in LDS per step (double buffered)

typedef __attribute__((ext_vector_type(2))) float v2f;
typedef __attribute__((ext_vector_type(8))) float v8f;

__device__ __forceinline__ v8f wmma_f32(v2f a, v2f b, v8f c) {
  // V_WMMA_F32_16X16X4_F32 : D(16x16,f32) = A(16x4,f32) x B(4x16,f32) + C
  return __builtin_amdgcn_wmma_f32_16x16x4_f32(
      /*neg_a=*/false, a, /*neg_b=*/false, b,
      /*c_mod=*/(short)0, c, /*reuse_a=*/false, /*reuse_b=*/false);
}

__global__ __launch_bounds__(BLOCK)
void siren_experts_kernel(const float* __restrict__ x,
                          const float* __restrict__ w0, const float* __restrict__ b0,
                          const float* __restrict__ w1, const float* __restrict__ b1,
                          const float* __restrict__ w2, const float* __restrict__ b2,
                          const float* __restrict__ w3, const float* __restrict__ b3,
                          const float* __restrict__ wf, const float* __restrict__ bf,
                          float* __restrict__ out)
{
  // Activation tile, row-major (32 KB)
  __shared__ float h_lds[ROWS][HID];
  // Double-buffered W^T panels in K-PAIR-interleaved layout (2 x 8 KB):
  // wp[buf][kp][o] = { W[o][kb+2*kp], W[o][kb+2*kp+1] }  -> B frag = 1x ds_load_b64
  __shared__ v2f  wp[2][KBLK / 2][HID];
  __shared__ float v_lds[HID * IN_DIM];     // w0 stage, later reused for wf (3 KB)
  __shared__ float x_lds[ROWS * IN_DIM];

  const int tid  = threadIdx.x;
  const int m    = blockIdx.x;              // expert
  const int rb   = blockIdx.y * ROWS;       // batch row base
  const int lane = tid & 31;
  const int wave = tid >> 5;
  const int n    = lane & 15;               // tile col (B/D) == tile row (A)
  const int half = lane >> 4;               // 0: lanes 0-15, 1: lanes 16-31

  // ---------------- stage x tile and w0 ----------------
  for (int p = tid; p < ROWS * IN_DIM; p += BLOCK) x_lds[p] = x[rb * IN_DIM + p];
  for (int p = tid; p < HID * IN_DIM;  p += BLOCK) v_lds[p] = w0[m * HID * IN_DIM + p];
  __syncthreads();

  // ---------------- layer 0 (K=3, VALU): h = sin(30*(x @ w0^T + b0)) --------
  for (int p = tid; p < ROWS * HID; p += BLOCK) {
    const int r = p >> 8;                   // HID == 256
    const int o = p & (HID - 1);
    float s = x_lds[r*3+0] * v_lds[o*3+0]
            + x_lds[r*3+1] * v_lds[o*3+1]
            + x_lds[r*3+2] * v_lds[o*3+2]
            + b0[m * HID + o];
    h_lds[r][o] = __sinf(W0S * s);
  }

  // ---------------- hidden layers (WMMA f32 16x16x4) ------------------------
  const float* Ws[3] = { w1, w2, w3 };
  const float* Bs[3] = { b1, b2, b3 };

  for (int layer = 0; layer < 3; ++layer) {
    const float* __restrict__ wl = Ws[layer] + (size_t)m * HID * HID;
    const float* __restrict__ bl = Bs[layer] + m * HID;

    v8f acc[16];
    const v8f vzero = { 0.f, 0.f, 0.f, 0.f, 0.f, 0.f, 0.f, 0.f };
    #pragma unroll
    for (int t = 0; t < 16; ++t) acc[t] = vzero;

    // cooperative panel stage: KBLK x HID floats = 512 float4 loads
    auto stage_panel = [&](int kb, int buf) {
      #pragma unroll
      for (int it = 0; it < (KBLK * HID / 4) / BLOCK; ++it) {
        const int p = tid + it * BLOCK;
        const int o = p >> 1;               // output neuron (row of W)
        const int q = p & 1;                // which float4 of the 8 k-values
        const float4 v = *reinterpret_cast<const float4*>(wl + o * HID + kb + q * 4);
        v2f lo, hi;
        lo.x = v.x; lo.y = v.y;
        hi.x = v.z; hi.y = v.w;
        wp[buf][q * 2 + 0][o] = lo;         // one ds_store_b64
        wp[buf][q * 2 + 1][o] = hi;         // one ds_store_b64
      }
    };

    __syncthreads();                         // wp free (prev layer compute done)
    stage_panel(0, 0);

    const int nkb = HID / KBLK;              // 32 panels
    for (int kbi = 0; kbi < nkb; ++kbi) {
      const int buf = kbi & 1;
      const int kb  = kbi * KBLK;
      __syncthreads();                       // panel[buf] ready; panel[buf^1] free
      if (kbi + 1 < nkb) stage_panel(kb + KBLK, buf ^ 1);

      #pragma unroll
      for (int s = 0; s < KBLK / 4; ++s) {
        // A fragment (16x4 f32): lane = m + 16*(k/2), vgpr = k%2 -> one b64 load
        const v2f a = *reinterpret_cast<const v2f*>(
            &h_lds[wave * 16 + n][kb + 4 * s + 2 * half]);
        #pragma unroll
        for (int nt = 0; nt < 16; ++nt) {
          // B fragment (4x16 f32): lane = n + 16*(k/2), vgpr = k%2 -> one b64 load
          const v2f b = wp[buf][2 * s + half][nt * 16 + n];
          acc[nt] = wmma_f32(a, b, acc[nt]);
        }
      }
    }

    // epilogue: h = sin(30*(acc + bias)), back to row-major.
    // C/D layout: vgpr r, lane -> (row = r + 8*half, col = n)
    #pragma unroll
    for (int nt = 0; nt < 16; ++nt) {
      const float bias = bl[nt * 16 + n];
      #pragma unroll
      for (int r = 0; r < 8; ++r) {
        h_lds[wave * 16 + 8 * half + r][nt * 16 + n] =
            __sinf(W0S * (acc[nt][r] + bias));
      }
    }
  }

  // ---------------- final layer (N=3, VALU): out = h @ wf^T + bf ------------
  __syncthreads();
  for (int p = tid; p < OUT_DIM * HID; p += BLOCK) v_lds[p] = wf[m * OUT_DIM * HID + p];
  __syncthreads();

  for (int p = tid; p < ROWS * OUT_DIM; p += BLOCK) {
    const int r = p / 3;
    const int o = p - r * 3;
    const int start = (tid * 4) & (HID - 1);   // rotate start to spread LDS banks
    float s = 0.f;
    #pragma unroll 8
    for (int j = 0; j < HID; ++j) {
      const int i = (start + j) & (HID - 1);
      s += h_lds[r][i] * v_lds[o * HID + i];
    }
    out[((size_t)(rb + r) * M_EXPERTS + m) * OUT_DIM + o] = s + bf[m * OUT_DIM + o];
  }
}

extern "C" void kernel_launch(void* const* d_in, const int* in_sizes, int n_in,
                              void* d_out, int out_size, void* d_ws, size_t ws_size,
                              hipStream_t stream) {
  const float* x  = (const float*)d_in[0];
  const float* w0 = (const float*)d_in[1];
  const float* b0 = (const float*)d_in[2];
  const float* w1 = (const float*)d_in[3];
  const float* b1 = (const float*)d_in[4];
  const float* w2 = (const float*)d_in[5];
  const float* b2 = (const float*)d_in[6];
  const float* w3 = (const float*)d_in[7];
  const float* b3 = (const float*)d_in[8];
  const float* wf = (const float*)d_in[9];
  const float* bf = (const float*)d_in[10];
  float* out = (float*)d_out;

  dim3 grid(M_EXPERTS, B_TOTAL / ROWS);   // 64 experts x 256 batch tiles
  siren_experts_kernel<<<grid, BLOCK, 0, stream>>>(
      x, w0, b0, w1, b1, w2, b2, w3, b3, wf, bf, out);
}